// localCrossCorrelation_48567490183745
// MI455X (gfx1250) — compile-verified
//
#include <hip/hip_runtime.h>
#include <hip/hip_bf16.h>
#include <stdint.h>

// LNCC loss, win=9 box filter along W only, images 16x1x1024x1024 fp32.
// Memory-bound: 128 MiB in / 4 B out -> HBM roofline ~5.6us on MI455X.
// Strategy: per-row LDS staging via CDNA5 async global->LDS copies
// (ASYNCcnt path), register-resident sliding-window sums, hierarchical
// reduction (wave32 shuffles -> LDS -> f64 global atomic).

#define ROW_W   1024
#define WIN     9
#define RAD     4
#define WPAD    (ROW_W + 2 * RAD)   // 1032 floats
#define EPS_F   1e-5f

// ---- CDNA5 async global->LDS helpers (gfx1250) -----------------------------
__device__ __forceinline__ void async_load_b128_to_lds(const float* gaddr,
                                                       uint32_t lds_byte_off) {
  // VDST = per-lane LDS byte offset, VADDR = 64-bit global address, GV mode.
  asm volatile("global_load_async_to_lds_b128 %0, %1, off"
               :
               : "v"(lds_byte_off), "v"(gaddr)
               : "memory");
}

__device__ __forceinline__ void wait_asynccnt0() {
#if defined(__has_builtin)
#if __has_builtin(__builtin_amdgcn_s_wait_asynccnt)
  __builtin_amdgcn_s_wait_asynccnt(0);
  return;
#endif
#endif
  asm volatile("s_wait_asynccnt 0x0" ::: "memory");
}

// LDS byte offset of a __shared__ object: addrspacecast(AS3->flat) keeps the
// LDS offset in the low 32 bits on AMDGPU.
__device__ __forceinline__ uint32_t lds_offset(const void* p) {
  return (uint32_t)(uintptr_t)p;
}

// ---- hot kernel (defined first so it leads the disassembly) ----------------
__global__ void __launch_bounds__(256)
lncc_row_kernel(const float* __restrict__ I, const float* __restrict__ J,
                double* __restrict__ acc) {
  __shared__ __align__(16) float sI[WPAD];
  __shared__ __align__(16) float sJ[WPAD];
  __shared__ float wsum[8];

  const int tid = threadIdx.x;                 // 0..255
  const long long row = blockIdx.x;            // one image row per block
  const long long base = row * (long long)ROW_W;

  // Zero halo (conv 'same' zero padding outside the row).
  if (tid < RAD) {
    sI[tid] = 0.0f;               sJ[tid] = 0.0f;
    sI[ROW_W + RAD + tid] = 0.0f; sJ[ROW_W + RAD + tid] = 0.0f;
  }

  // Stage the full 1024-float row of each image into LDS: one async b128
  // (4 floats) per thread per image. 16B-aligned on both sides.
  const int c = tid * 4;                       // first pixel this thread owns
  async_load_b128_to_lds(I + base + c, lds_offset(&sI[RAD + c]));
  async_load_b128_to_lds(J + base + c, lds_offset(&sJ[RAD + c]));
  wait_asynccnt0();
  __syncthreads();

  // Pull this thread's 12-float sliding footprint into registers with three
  // aligned ds_load_b128 per array (sX[c .. c+11], c % 4 == 0).
  float a[12], b[12];
  {
    const float4* vI = (const float4*)(&sI[c]);
    const float4* vJ = (const float4*)(&sJ[c]);
    float4 i0 = vI[0], i1 = vI[1], i2 = vI[2];
    float4 j0 = vJ[0], j1 = vJ[1], j2 = vJ[2];
    a[0]=i0.x; a[1]=i0.y; a[2]=i0.z; a[3]=i0.w;
    a[4]=i1.x; a[5]=i1.y; a[6]=i1.z; a[7]=i1.w;
    a[8]=i2.x; a[9]=i2.y; a[10]=i2.z; a[11]=i2.w;
    b[0]=j0.x; b[1]=j0.y; b[2]=j0.z; b[3]=j0.w;
    b[4]=j1.x; b[5]=j1.y; b[6]=j1.z; b[7]=j1.w;
    b[8]=j2.x; b[9]=j2.y; b[10]=j2.z; b[11]=j2.w;
  }

  // Sliding-window sums for 4 adjacent pixels; window for output pixel
  // x = c + j is elements [j .. j+8] of the register footprint.
  float SI = 0.f, SJ = 0.f, SI2 = 0.f, SJ2 = 0.f, SIJ = 0.f;
#pragma unroll
  for (int t = 0; t < WIN; ++t) {
    float iv = a[t], jv = b[t];
    SI += iv; SJ += jv;
    SI2 += iv * iv; SJ2 += jv * jv; SIJ += iv * jv;
  }

  const float inv_w = 1.0f / (float)WIN;
  float accv = 0.0f;
#pragma unroll
  for (int j = 0; j < 4; ++j) {
    // cross = IJs - Is*Js/9 ; var = X2s - Xs^2/9  (algebraically identical
    // to the reference's u_I/u_J formulation)
    float cross = SIJ - SI * SJ * inv_w;
    float varI  = SI2 - SI * SI * inv_w;
    float varJ  = SJ2 - SJ * SJ * inv_w;
    accv += (cross * cross) / (varI * varJ + EPS_F);
    if (j < 3) {
      float ri = a[j],       rj = b[j];        // element leaving the window
      float ai = a[j + WIN], aj = b[j + WIN];  // element entering the window
      SI  += ai - ri;            SJ  += aj - rj;
      SI2 += ai * ai - ri * ri;  SJ2 += aj * aj - rj * rj;
      SIJ += ai * aj - ri * rj;
    }
  }

  // Reduce 256 threads -> 1 value. wave32 on gfx1250.
#pragma unroll
  for (int off = 16; off > 0; off >>= 1)
    accv += __shfl_xor(accv, off, 32);
  const int wave = tid >> 5, lane = tid & 31;
  if (lane == 0) wsum[wave] = accv;
  __syncthreads();
  if (tid == 0) {
    float s = 0.0f;
#pragma unroll
    for (int w = 0; w < 8; ++w) s += wsum[w];
    atomicAdd(acc, (double)s);   // global_atomic_add_f64
  }
}

// ---- small kernels ---------------------------------------------------------
__global__ void lncc_init_kernel(double* acc) {
  if (blockIdx.x == 0 && threadIdx.x == 0) acc[0] = 0.0;
}

__global__ void lncc_finalize_kernel(const double* __restrict__ acc,
                                     float* __restrict__ out,
                                     double inv_npix) {
  if (blockIdx.x == 0 && threadIdx.x == 0) {
    out[0] = (float)(1.0 - acc[0] * inv_npix);
  }
}

// ---- entry -----------------------------------------------------------------
extern "C" void kernel_launch(void* const* d_in, const int* in_sizes, int n_in,
                              void* d_out, int out_size, void* d_ws, size_t ws_size,
                              hipStream_t stream) {
  const float* I = (const float*)d_in[0];
  const float* J = (const float*)d_in[1];
  float* out = (float*)d_out;
  double* acc = (double*)d_ws;

  const long long total = (long long)in_sizes[0];   // 16*1024*1024
  const int rows = (int)(total / ROW_W);            // 16384
  const double inv_npix = 1.0 / (double)total;

  lncc_init_kernel<<<1, 1, 0, stream>>>(acc);
  lncc_row_kernel<<<rows, 256, 0, stream>>>(I, J, acc);
  lncc_finalize_kernel<<<1, 1, 0, stream>>>(acc, out, inv_npix);
}